// Utils_16956530884645
// MI455X (gfx1250) — compile-verified
//
#include <hip/hip_runtime.h>
#include <math.h>

// MI455X sphere-sweep warp.
// Roofline: 402 MB output / 23.3 TB/s ~= 17 us floor; inputs (12 MB) live in
// the 192 MB L2, so gathers are cache hits. Stores are nontemporal to keep
// the input images resident. Rotation of unit directions is done with
// V_WMMA_F32_16X16X4_F32 (the one matmul-shaped stage).
// Hot-loop transcendentals use raw HW approx ops (v_rcp/v_rsq/v_sqrt), not
// the OCML IEEE-divide path the previous round emitted (v_div_scale chains).

typedef __attribute__((ext_vector_type(2))) float v2f;
typedef __attribute__((ext_vector_type(8))) float v8f;

#define WIDTH   1024
#define HEIGHT  512
#define DEPTH   32
#define HW      (HEIGHT * WIDTH)
#define TWO_PI_F 6.28318530717958647692f
#define PI_F     3.14159265358979323846f

// Branch-free acos, abs err ~2e-8 (classic sqrt(1-x)*poly form).
__device__ __forceinline__ float fast_acosf(float x) {
    float ax = fabsf(x);
    float p = -0.0012624911f;
    p = fmaf(p, ax,  0.0066700901f);
    p = fmaf(p, ax, -0.0170881256f);
    p = fmaf(p, ax,  0.0308918810f);
    p = fmaf(p, ax, -0.0501743046f);
    p = fmaf(p, ax,  0.0889789874f);
    p = fmaf(p, ax, -0.2145988016f);
    p = fmaf(p, ax,  1.5707963050f);
    float r = p * __builtin_amdgcn_sqrtf(fmaxf(1.0f - ax, 0.0f));
    return (x < 0.0f) ? (PI_F - r) : r;
}

// Branch-free atan2, err ~1e-6 rad: min/max ratio (single v_rcp_f32) +
// odd minimax poly + quadrant fixups.
__device__ __forceinline__ float fast_atan2f(float y, float x) {
    float ax = fabsf(x), ay = fabsf(y);
    float mx = fmaxf(ax, ay), mn = fminf(ax, ay);
    float a  = mn * __builtin_amdgcn_rcpf(mx);   // one v_rcp_f32, ~1 ulp
    float s  = a * a;
    float p = -0.01172120f;
    p = fmaf(p, s,  0.05265332f);
    p = fmaf(p, s, -0.11643287f);
    p = fmaf(p, s,  0.19354346f);
    p = fmaf(p, s, -0.33262347f);
    p = fmaf(p, s,  0.99997726f);
    float r = p * a;
    r = (ay > ax)   ? (1.57079632679489662f - r) : r;
    r = (x < 0.0f)  ? (PI_F - r) : r;
    r = (y < 0.0f)  ? -r : r;
    return r;
}

__global__ __launch_bounds__(256) void sphere_sweep_warp(
    const float* __restrict__ view_ref,  // [3,H,W]
    const float* __restrict__ view_inp,  // [3,H,W]
    const float* __restrict__ r_ref,     // [3,3]
    const float* __restrict__ t_ref,     // [3]
    const float* __restrict__ r_inp,     // [3,3]
    const float* __restrict__ t_inp,     // [3]
    const float* __restrict__ radii,     // [D]
    float* __restrict__ out)             // [D,6,H,W]
{
    const int tid = threadIdx.x;
    const int w   = blockIdx.x * blockDim.x + tid;
    const int h   = blockIdx.y;
    const int v   = blockIdx.z;                       // 0 = ref, 1 = input

    const float* __restrict__ img = v ? view_inp : view_ref;
    const float* __restrict__ Rm  = v ? r_inp    : r_ref;
    const float* __restrict__ tv  = v ? t_inp    : t_ref;
    const int cb = v * 3;                             // output channel base

    __shared__ float sRad[DEPTH];
    if (tid < DEPTH) sRad[tid] = radii[tid];
    __syncthreads();

    // ---- per-pixel unit direction (equi_2_spherical -> cartesian) ----
    float theta = ((float)w + 0.5f) * (TWO_PI_F / (float)WIDTH);
    float phi   = ((float)h + 0.5f) * (PI_F     / (float)HEIGHT);
    float sp, cp, st, ct;
    sincosf(phi, &sp, &cp);
    sincosf(theta, &st, &ct);
    float dx = sp * ct, dy = sp * st, dz = cp;

    // ---- rd = R * dir via V_WMMA_F32_16X16X4_F32 (ISA 7.12.2 layouts) ----
    // A (16x4): rows 0-2 and 8-10 hold R (duplicate so lanes 16-31 of D are
    // valid); lane L holds A[L&15][k], k = 2*(L>>4) in VGPR0, k+1 in VGPR1.
    const int lane = tid & 31;                        // wave32
    const int m    = lane & 15;
    const int kb   = (lane >> 4) << 1;                // 0 or 2
    const int row  = m & 7;
    const bool valid = (row < 3);
    v2f A;
    A.x = valid ? Rm[row * 3 + kb] : 0.0f;            // k = 0 or 2 (<3)
    A.y = (valid && kb == 0) ? Rm[row * 3 + 1] : 0.0f;// k = 1 valid, k = 3 -> 0

    // B (4x16): column n = dir of lane n; lane L supplies column L&15 with
    // K = 0,1 (lanes 0-15) or K = 2,3 (lanes 16-31). Two WMMAs cover the
    // wave's 32 pixels (WMMA1: lanes 0-15, WMMA2: lanes 16-31).
    const int s1 = lane & 15;
    const int s2 = s1 + 16;
    float d0a = __shfl(dx, s1, 32), d1a = __shfl(dy, s1, 32), d2a = __shfl(dz, s1, 32);
    float d0b = __shfl(dx, s2, 32), d1b = __shfl(dy, s2, 32), d2b = __shfl(dz, s2, 32);
    const bool hi = lane >= 16;
    v2f B1; B1.x = hi ? d2a : d0a; B1.y = hi ? 0.0f : d1a;
    v2f B2; B2.x = hi ? d2b : d0b; B2.y = hi ? 0.0f : d1b;

    v8f cz = {};
    v8f D1 = __builtin_amdgcn_wmma_f32_16x16x4_f32(false, A, false, B1,
                                                   (short)0, cz, false, false);
    v8f D2 = __builtin_amdgcn_wmma_f32_16x16x4_f32(false, A, false, B2,
                                                   (short)0, cz, false, false);
    // Lane L (<16): D1 VGPR j -> M=j, N=L.  Lane L (>=16): D2 VGPR j -> M=8+j,
    // N=L-16 (duplicated R rows) == rd of pixel L. No post-WMMA shuffles.
    float rdx = hi ? D2[0] : D1[0];
    float rdy = hi ? D2[1] : D1[1];
    float rdz = hi ? D2[2] : D1[2];

    const float tx = tv[0], ty = tv[1], tz = tv[2];
    const float* __restrict__ p0 = img;
    const float* __restrict__ p1 = img + HW;
    const float* __restrict__ p2 = img + 2 * HW;

    #pragma unroll 4
    for (int d = 0; d < DEPTH; ++d) {
        float r  = sRad[d];
        float px = fmaf(r, rdx, tx);
        float py = fmaf(r, rdy, ty);
        float pz = fmaf(r, rdz, tz);

        // atan2 needs no normalization; only nz does. Raw v_rsq_f32 is ample.
        float inv = __builtin_amdgcn_rsqf(fmaf(px, px, fmaf(py, py, pz * pz)));
        float nz  = fminf(fmaxf(pz * inv, -1.0f), 1.0f);
        float ph  = fast_acosf(nz);
        float th  = fast_atan2f(py, px);
        th = (th < 0.0f) ? th + TWO_PI_F : th;        // jnp.mod(.., 2pi)

        float gx = fmaf(th, (float)WIDTH  / TWO_PI_F, -0.5f);
        float gy = fmaf(ph, (float)HEIGHT / PI_F,     -0.5f);
        float fx = floorf(gx), fy = floorf(gy);
        float wx = gx - fx,    wy = gy - fy;

        int xi = (int)fx, yi = (int)fy;
        int x0 = xi & (WIDTH - 1);                    // wrap (W = 2^10)
        int x1 = (xi + 1) & (WIDTH - 1);
        int y0 = yi < 0 ? 0 : (yi > HEIGHT - 1 ? HEIGHT - 1 : yi);
        int y1 = (yi + 1 > HEIGHT - 1) ? HEIGHT - 1 : (yi + 1 < 0 ? 0 : yi + 1);

        int i00 = y0 * WIDTH + x0, i01 = y0 * WIDTH + x1;
        int i10 = y1 * WIDTH + x0, i11 = y1 * WIDTH + x1;
        float w00 = (1.0f - wx) * (1.0f - wy);
        float w01 = wx * (1.0f - wy);
        float w10 = (1.0f - wx) * wy;
        float w11 = wx * wy;

        float* o = out + ((size_t)d * 6 + cb) * HW + (size_t)h * WIDTH + w;

        float c0v = p0[i00]*w00 + p0[i01]*w01 + p0[i10]*w10 + p0[i11]*w11;
        float c1v = p1[i00]*w00 + p1[i01]*w01 + p1[i10]*w10 + p1[i11]*w11;
        float c2v = p2[i00]*w00 + p2[i01]*w01 + p2[i10]*w10 + p2[i11]*w11;

        // Nontemporal: keep the 12 MB of input images resident in L2 while
        // streaming the 402 MB output.
        __builtin_nontemporal_store(c0v, o);
        __builtin_nontemporal_store(c1v, o + HW);
        __builtin_nontemporal_store(c2v, o + 2 * HW);
    }
}

extern "C" void kernel_launch(void* const* d_in, const int* in_sizes, int n_in,
                              void* d_out, int out_size, void* d_ws, size_t ws_size,
                              hipStream_t stream) {
    (void)in_sizes; (void)n_in; (void)out_size; (void)d_ws; (void)ws_size;
    const float* view_ref = (const float*)d_in[0];
    const float* view_inp = (const float*)d_in[1];
    const float* r_ref    = (const float*)d_in[2];
    const float* t_ref    = (const float*)d_in[3];
    const float* r_inp    = (const float*)d_in[4];
    const float* t_inp    = (const float*)d_in[5];
    const float* radii    = (const float*)d_in[6];
    float* out = (float*)d_out;

    dim3 grid(WIDTH / 256, HEIGHT, 2);   // (w tiles, h rows, view)
    sphere_sweep_warp<<<grid, 256, 0, stream>>>(
        view_ref, view_inp, r_ref, t_ref, r_inp, t_inp, radii, out);
}